// RoIAlign_80109730005434
// MI455X (gfx1250) — compile-verified
//
#include <hip/hip_runtime.h>
#include <cstdint>
#include <cstddef>

// ---------------- problem constants (from reference) ----------------
#define OUT_HW   7
#define C_TOT    256
#define H_F      200
#define W_F      200
#define CHUNK_C  64                         // channels per staged LDS chunk
#define NCHUNK   (C_TOT / CHUNK_C)          // 4
#define MAX_COLS 65                         // span<=63 -> cols=span+2<=65
#define BUF_ELEMS (CHUNK_C * 2 * MAX_COLS)  // 8320 floats per buffer
#define STAGE_ITERS 33                      // fallback path: ceil(64*2*65/256)

// ---------------- CDNA5 feature detection ----------------
#if defined(__HIP_DEVICE_COMPILE__) && \
    __has_builtin(__builtin_amdgcn_tensor_load_to_lds) && \
    __has_builtin(__builtin_amdgcn_s_wait_tensorcnt)
#define USE_TDM 1
#else
#define USE_TDM 0
#endif

#if defined(__HIP_DEVICE_COMPILE__) && \
    __has_builtin(__builtin_amdgcn_global_load_async_to_lds_b32) && \
    __has_builtin(__builtin_amdgcn_s_wait_asynccnt)
#define USE_ASYNC_LDS 1
#else
#define USE_ASYNC_LDS 0
#endif

typedef uint32_t v4u __attribute__((ext_vector_type(4)));
typedef int      v8i __attribute__((ext_vector_type(8)));
typedef int      v4i __attribute__((ext_vector_type(4)));
typedef __attribute__((address_space(1))) int* gptr_i32;
typedef __attribute__((address_space(3))) int* lptr_i32;

__device__ __forceinline__ uint32_t lds_byte_addr(void* p) {
    return (uint32_t)(uintptr_t)(__attribute__((address_space(3))) void*)p;
}

__device__ __forceinline__ void async_cp_b32(float* dst_lds, const float* src_glb) {
#if USE_ASYNC_LDS
    __builtin_amdgcn_global_load_async_to_lds_b32(
        (gptr_i32)const_cast<float*>(src_glb), (lptr_i32)dst_lds, /*off*/0, /*cpol*/0);
#else
    *dst_lds = __ldg(src_glb);
#endif
}

template <int N>
__device__ __forceinline__ void wait_async_lds() {
#if USE_ASYNC_LDS
    __builtin_amdgcn_s_wait_asynccnt(N);
#endif
}

#if USE_TDM
// One TDM descriptor stages a 3D tile: x (colsPad contiguous) * y (2 rows,
// stride W) * c (64 channels, stride H*W) -> packed LDS block.
__device__ __forceinline__ void tdm_load_tile(float* lds_dst, const float* gsrc, int colsPad) {
    uint32_t la    = lds_byte_addr(lds_dst);
    uint64_t ga    = (uint64_t)(uintptr_t)gsrc;
    uint32_t ga_lo = (uint32_t)ga;
    uint32_t ga_hi = (uint32_t)(ga >> 32);
    // Descriptor lives in SGPRs -> force uniform.
    la    = (uint32_t)__builtin_amdgcn_readfirstlane((int)la);
    ga_lo = (uint32_t)__builtin_amdgcn_readfirstlane((int)ga_lo);
    ga_hi = (uint32_t)__builtin_amdgcn_readfirstlane((int)ga_hi);
    int cp = __builtin_amdgcn_readfirstlane(colsPad);

    // D# group 0: count=1 | lds_addr | global_addr[56:0] | type=2
    v4u g0 = { 1u, la, ga_lo, ga_hi | (2u << 30) };
    // D# group 1: mask=0, data_size=2(4B); tensor_dim0/1=200; tile=(cp,2,64);
    //             dim0_stride=200, dim1_stride=40000
    v8i g1;
    g1[0] = 0x20000;                                   // data_size=4B, no flags
    g1[1] = (W_F & 0xFFFF) << 16;                      // abar=0 | tensor_dim0[15:0]
    g1[2] = (H_F & 0xFFFF) << 16;                      // td0 hi=0 | tensor_dim1[15:0]
    g1[3] = (cp & 0xFFFF) << 16;                       // td1 hi=0 | tile_dim0
    g1[4] = 2 | (CHUNK_C << 16);                       // tile_dim1=2 | tile_dim2=64
    g1[5] = W_F;                                       // tensor_dim0_stride lo32
    g1[6] = (int)(((uint32_t)(H_F * W_F) & 0xFFFFu) << 16); // s0 hi=0 | s1[15:0]
    g1[7] = (int)((uint32_t)(H_F * W_F) >> 16);        // s1[47:16]
    v4i g2 = { C_TOT, 0, 0, 0 };                       // tensor_dim2=256, no dim3
    v4i g3 = { 0, 0, 0, 0 };
#if __clang_major__ >= 23
    v8i g4 = { 0, 0, 0, 0, 0, 0, 0, 0 };
    __builtin_amdgcn_tensor_load_to_lds(g0, g1, g2, g3, g4, 0);
#else
    __builtin_amdgcn_tensor_load_to_lds(g0, g1, g2, g3, 0);
#endif
}
#endif

// ---------------- kernel: one block per (roi, oy) ----------------
__global__ __launch_bounds__(256) void roialign_cdna5_kernel(
    const float* __restrict__ feat,   // (4,256,200,200)
    const int*   __restrict__ rois,   // (N,5) = b,x1,y1,x2,y2
    float*       __restrict__ out)    // (N,256,7,7)
{
    __shared__ float smem[2][BUF_ELEMS + 1];   // +1 = dump slot (fallback pad)

    const int bx = blockIdx.x;
    const int n  = bx / OUT_HW;
    const int oy = bx - n * OUT_HW;
    const int t  = threadIdx.x;

    const int b   = rois[n * 5 + 0];
    const int rx1 = rois[n * 5 + 1];
    const int ry1 = rois[n * 5 + 2];
    const int rx2 = rois[n * 5 + 3];
    const int ry2 = rois[n * 5 + 4];

    // y sampling (matches reference fp32 order: mul then /6; weight from
    // unclamped floor)
    const float sy  = (float)ry1 + ((float)oy * (float)(ry2 - ry1)) / 6.0f;
    const float y0f = floorf(sy);
    const float wy  = sy - y0f;
    int y0 = min(max((int)y0f, 0), H_F - 2);   // y0+1 always in range (y2<=198)

    const int span    = rx2 - rx1;             // 8..63 (never clipped)
    const int cols    = span + 2;              // needed columns, <= 65
    const int colsPad = cols | 1;              // odd -> stride 2*colsPad == 2 mod 4
    const int stride  = 2 * colsPad;           //   -> conflict-free LDS channel stride
    const int total   = CHUNK_C * 2 * cols;    // fallback: staged elements/chunk

    const float* fB    = feat + (size_t)b * C_TOT * H_F * W_F;
    const float* tile0 = fB + (size_t)y0 * W_F + rx1;   // chunk c0: + c0*H*W

#if !USE_TDM
    // Fallback staging: all 8 waves issue exactly STAGE_ITERS async b32 copies
    // (padding goes to dump slot so wait thresholds are compile-time imms).
    auto stage = [&](int chunk, int bufsel) {
        const int c0 = chunk * CHUNK_C;
#pragma unroll 1
        for (int i = 0; i < STAGE_ITERS; ++i) {
            int e        = t + i * 256;
            int valid    = (e < total);
            unsigned ee  = valid ? (unsigned)e : 0u;
            unsigned row = ee / (unsigned)cols;     // (c<<1)|r
            unsigned xi  = ee - row * (unsigned)cols;
            int c = (int)(row >> 1);
            int r = (int)(row & 1u);
            const float* src = fB + ((size_t)(c0 + c) * H_F + (y0 + r)) * W_F + rx1 + (int)xi;
            float* dst = &smem[bufsel][valid ? (c * stride + r * colsPad + (int)xi)
                                             : BUF_ELEMS];
            async_cp_b32(dst, src);
        }
    };
#endif

    const bool issuer = (t < 32);              // wave 0 drives the TDM

#if USE_TDM
    if (issuer) tdm_load_tile(&smem[0][0], tile0, colsPad);
#else
    stage(0, 0);
#endif

    const float x1f  = (float)rx1;
    const float dxf  = (float)(rx2 - rx1);
    const float omwy = 1.0f - wy;

    for (int k = 0; k < NCHUNK; ++k) {
#if USE_TDM
        if (issuer) {
            if (k < NCHUNK - 1) {
                tdm_load_tile(&smem[(k + 1) & 1][0],
                              tile0 + (size_t)((k + 1) * CHUNK_C) * (H_F * W_F), colsPad);
                __builtin_amdgcn_s_wait_tensorcnt(1);  // in-order: chunk k landed
            } else {
                __builtin_amdgcn_s_wait_tensorcnt(0);
            }
        }
#else
        if (k < NCHUNK - 1) {
            stage(k + 1, (k + 1) & 1);
            wait_async_lds<STAGE_ITERS>();
        } else {
            wait_async_lds<0>();
        }
#endif
        __syncthreads();                        // chunk k visible to all waves

        const float* B  = smem[k & 1];
        const int    c0 = k * CHUNK_C;

#pragma unroll
        for (int j = 0; j < 2; ++j) {
            int w = t + j * 256;                // work item over 7*64 = 448
            if (w < OUT_HW * CHUNK_C) {
                int c  = w & (CHUNK_C - 1);     // lanes over channels
                int ox = w >> 6;

                float sx  = x1f + ((float)ox * dxf) / 6.0f;
                float x0f = floorf(sx);
                float wx  = sx - x0f;
                int   x0  = min(max((int)x0f, 0), W_F - 1);
                int   xi  = x0 - rx1;           // 0..span; xi+1 <= cols-1

                const float* L = B + c * stride;
                float v00 = L[xi];
                float v01 = L[xi + 1];
                float v10 = L[colsPad + xi];
                float v11 = L[colsPad + xi + 1];

                float acc;
                {
#pragma clang fp contract(off)
                    float omwx = 1.0f - wx;
                    acc = v00 * omwy * omwx + v01 * omwy * wx
                        + v10 * wy * omwx + v11 * wy * wx;
                }

                float* op = out + (((size_t)(n * C_TOT + c0 + c) * OUT_HW) + oy) * OUT_HW + ox;
                __builtin_nontemporal_store(acc, op);  // keep features L2-resident
            }
        }
        __syncthreads();                        // WAR: buffer reused 2 iters later
    }
}

// ---------------- launcher ----------------
extern "C" void kernel_launch(void* const* d_in, const int* in_sizes, int n_in,
                              void* d_out, int out_size, void* d_ws, size_t ws_size,
                              hipStream_t stream) {
    const float* feat = (const float*)d_in[0];
    const int*   rois = (const int*)d_in[1];
    float*       out  = (float*)d_out;

    const int N = in_sizes[1] / 5;              // 512
    dim3 grid(N * OUT_HW);                      // 3584 blocks, 256 threads (8 waves)
    roialign_cdna5_kernel<<<grid, 256, 0, stream>>>(feat, rois, out);
}